// MLA_1357209666321
// MI455X (gfx1250) — compile-verified
//
#include <hip/hip_runtime.h>
#include <hip/hip_bf16.h>

// ---------------------------------------------------------------------------
// MLA decode (DeepSeek-style) for MI455X / gfx1250.
// Bandwidth-bound: kv_cache (268MB fp32) read exactly once via fused
// flash-decoding chunks held in 320KB-per-WGP LDS. Matrix work uses
// v_wmma_f32_16x16x4_f32 (M=16 == n_heads fits one WMMA tile); the chunk
// global->LDS transfer uses the CDNA5 Tensor Data Mover: each pass-1 wave
// issues TDM descriptors for ITS OWN 16-row tile and self-synchronizes on
// its per-wave TENSORcnt, so WMMA compute on early tiles overlaps the DMA
// of later tiles (no workgroup-wide load barrier).
// ---------------------------------------------------------------------------

typedef float v2f __attribute__((ext_vector_type(2)));
typedef float v8f __attribute__((ext_vector_type(8)));
typedef unsigned int v4u __attribute__((ext_vector_type(4)));
typedef int v4i __attribute__((ext_vector_type(4)));
typedef int v8i __attribute__((ext_vector_type(8)));

#define N_HEADS 16
#define NOPE    128
#define ROPE    64
#define V_DIM   128
#define KV_RANK 512
#define KDIM    576          // KV_RANK + ROPE
#define SEQ     8193         // end_pos (start_pos 8192 + 1)
#define CHUNK   96           // positions per flash chunk (LDS-resident)
#define NCH     86           // ceil(8193/96)
#define KSTR    580          // kv LDS row stride in dwords (512 + 2x34 TDM pad)
#define ESTR    68           // pe LDS row stride in dwords (64 + 4 TDM pad)
#define PSTR    100          // LDS prob-row stride
#define SCALE   0.07216878364870323f   // (NOPE+ROPE)^-0.5 = 192^-0.5

// LDS layout of the attention kernel (float offsets into dynamic LDS)
#define L_KV    0                       // 96 x 580
#define L_PE    (CHUNK * KSTR)          // 96 x 68   -> 55680
#define L_Q     (L_PE + CHUNK * ESTR)   // 16 x 580  -> 62208
#define L_P     (L_Q + 16 * KSTR)       // 16 x 100  -> 71488
#define L_TOT   (L_P + 16 * PSTR)       // 73088 floats = 292,352 B

// ---- workspace layout (in floats) ----
#define OFF_QRAW   ((size_t)0)          // 16 x 3648  (q 3072 | kv_full 576)
#define OFF_QCAT   ((size_t)58368)      // 16 x 16 x 576  (q_abs | roped q_pe)
#define OFF_NKV    ((size_t)205824)     // 16 x 512   (rms-normed new kv row)
#define OFF_NPE    ((size_t)214016)     // 16 x 64    (roped new k_pe row)
#define OFF_M      ((size_t)215040)     // 16 x 86 x 16   chunk max
#define OFF_L      ((size_t)237056)     // 16 x 86 x 16   chunk sumexp
#define OFF_O      ((size_t)259072)     // 16 x 86 x 16 x 512 chunk O~
#define OFF_OHEAD  ((size_t)11531264)   // 16 x 16 x 512
#define OFF_OFLAT  ((size_t)11662336)   // 16 x 2048
// total = 11,695,104 floats = ~46.8 MB of d_ws

static __device__ __forceinline__ v8f wmma4(v2f a, v2f b, v8f c) {
  // D = A(16x4 f32) * B(4x16 f32) + C(16x16 f32)
  return __builtin_amdgcn_wmma_f32_16x16x4_f32(
      /*neg_a=*/false, a, /*neg_b=*/false, b,
      /*c_mod=*/(short)0, c, /*reuse_a=*/false, /*reuse_b=*/false);
}

// ---------------------------------------------------------------------------
// Issue one 2D Tensor-Data-Mover load: rows x width_dw f32 tile from global
// into LDS, with pad_amount dwords of LDS padding inserted after every
// 2^(pad_code+1) dwords (hardware swizzle for bank-conflict-free reads).
// Rows beyond rows_avail are zero-filled by TDM OOB semantics.
// D# layout per CDNA5 ISA ch. 8.3/8.4 (group0: count/lds/global/type,
// group1: data_size|pad|dims|strides, groups 2-3 zero for 2D tensors).
// ---------------------------------------------------------------------------
static __device__ __forceinline__ void tdm_load_2d(unsigned lds_byte,
                                                   const void* gptr,
                                                   unsigned width_dw,
                                                   unsigned rows_avail,
                                                   unsigned tile_rows,
                                                   unsigned pad_code,
                                                   unsigned pad_amt_code) {
  const unsigned long long ga = (unsigned long long)(uintptr_t)gptr;
  v4u g0;
  g0[0] = 1u;                                     // count=1, user mode
  g0[1] = lds_byte;                               // lds_addr (bytes)
  g0[2] = (unsigned)(ga & 0xFFFFFFFFu);           // global_addr[31:0]
  g0[3] = (unsigned)((ga >> 32) & 0x01FFFFFFu)    // global_addr[56:32]
          | (2u << 30);                           // type = 2 ("image")
  v8i g1;
  g1[0] = (int)((2u << 16)                        // data_size = 4B
                | (1u << 20)                      // pad_enable
                | (pad_code << 22)                // pad_interval
                | (pad_amt_code << 25));          // pad_amount
  g1[1] = (int)((width_dw & 0xFFFFu) << 16);      // tensor_dim0[15:0]
  g1[2] = (int)(((width_dw >> 16) & 0xFFFFu)      // tensor_dim0[31:16]
                | ((rows_avail & 0xFFFFu) << 16));// tensor_dim1[15:0]
  g1[3] = (int)(((rows_avail >> 16) & 0xFFFFu)    // tensor_dim1[31:16]
                | ((width_dw & 0xFFFFu) << 16));  // tile_dim0
  g1[4] = (int)(tile_rows & 0xFFFFu);             // tile_dim1 (tile_dim2=0)
  g1[5] = (int)width_dw;                          // tensor_dim0_stride[31:0]
  g1[6] = 0;
  g1[7] = 0;
  v4i gz = {0, 0, 0, 0};
#if defined(__clang_major__) && (__clang_major__ >= 23)
  v8i gz8 = {0, 0, 0, 0, 0, 0, 0, 0};
  __builtin_amdgcn_tensor_load_to_lds(g0, g1, gz, gz, gz8, 0);
#else
  __builtin_amdgcn_tensor_load_to_lds(g0, g1, gz, gz, 0);
#endif
}

// ---------------------------------------------------------------------------
// Generic row-GEMV, cols fixed at 2048. One wave per output row, lanes split
// the 2048-dim reduction with coalesced weight reads.
// ---------------------------------------------------------------------------
__global__ void gemv2048_kernel(const float* __restrict__ X,
                                const float* __restrict__ W,
                                float* __restrict__ Y,
                                int rows, int xstride, int ystride) {
  __shared__ float xs[2048];
  const int b = blockIdx.x, rb = blockIdx.y, tid = threadIdx.x;
  for (int i = tid; i < 2048; i += 256) xs[i] = X[(size_t)b * xstride + i];
  __syncthreads();
  const int wave = tid >> 5, lane = tid & 31;
  const int r0 = rb * 64 + wave * 8;
  for (int r = r0; r < r0 + 8; ++r) {
    if (r >= rows) break;
    const float* wr = W + (size_t)r * 2048;
    float s = 0.f;
    for (int j = lane; j < 2048; j += 32) s += wr[j] * xs[j];
    for (int off = 16; off > 0; off >>= 1) s += __shfl_xor(s, off, 32);
    if (lane == 0) Y[(size_t)b * ystride + r] = s;
  }
}

// ---------------------------------------------------------------------------
// Per-batch prep: RMS-norm new kv row, RoPE k_pe and q_pe, q_abs projection.
// New cache row goes to workspace (input caches stay read-only).
// ---------------------------------------------------------------------------
__global__ void prep_kernel(const float* __restrict__ qraw,
                            const float* __restrict__ kvnw,
                            const float* __restrict__ wkv_b,
                            const float* __restrict__ cosv,
                            const float* __restrict__ sinv,
                            float* __restrict__ qcat,
                            float* __restrict__ nkv,
                            float* __restrict__ npe) {
  __shared__ float qb[3648];
  __shared__ float red[256];
  const int b = blockIdx.x, tid = threadIdx.x;
  for (int i = tid; i < 3648; i += 256) qb[i] = qraw[(size_t)b * 3648 + i];
  __syncthreads();

  // RMS norm of kv (qb[3072..3583])
  float p = 0.f;
  for (int i = tid; i < 512; i += 256) { float v = qb[3072 + i]; p += v * v; }
  red[tid] = p; __syncthreads();
  for (int s = 128; s > 0; s >>= 1) {
    if (tid < s) red[tid] += red[tid + s];
    __syncthreads();
  }
  const float rscale = rsqrtf(red[0] * (1.0f / 512.0f) + 1e-6f);
  for (int i = tid; i < 512; i += 256)
    nkv[(size_t)b * 512 + i] = qb[3072 + i] * rscale * kvnw[i];

  // RoPE k_pe (qb[3584..3647], 32 complex pairs)
  if (tid < 32) {
    const int i = tid;
    const float c = cosv[i], s = sinv[i];
    const float xr = qb[3584 + 2 * i], xi = qb[3585 + 2 * i];
    npe[(size_t)b * 64 + 2 * i]     = xr * c - xi * s;
    npe[(size_t)b * 64 + 2 * i + 1] = xr * s + xi * c;
  }

  // RoPE q_pe for every head -> q_cat[...,512:576]
  for (int it = tid; it < 512; it += 256) {
    const int h = it >> 5, i = it & 31;
    const float c = cosv[i], s = sinv[i];
    const float xr = qb[h * 192 + NOPE + 2 * i];
    const float xi = qb[h * 192 + NOPE + 2 * i + 1];
    float* qr = qcat + ((size_t)(b * 16 + h)) * KDIM + KV_RANK;
    qr[2 * i]     = xr * c - xi * s;
    qr[2 * i + 1] = xr * s + xi * c;
  }

  // q_abs[h,c] = sum_d q_nope[h,d] * wkv_b[h*256+d, c]
  for (int o = tid; o < 16 * 512; o += 256) {
    const int h = o >> 9, cc = o & 511;
    const float* qn = qb + h * 192;
    const float* wb = wkv_b + ((size_t)h * 256) * 512 + cc;
    float s = 0.f;
    for (int d = 0; d < 128; ++d) s += qn[d] * wb[(size_t)d * 512];
    qcat[((size_t)(b * 16 + h)) * KDIM + cc] = s;
  }
}

// ---------------------------------------------------------------------------
// Flash-decoding chunk: one (chunk, batch) per workgroup, 256 threads.
// Waves 0..5 each TDM-load their own 16-row kc/pe tile (per-wave TENSORcnt
// self-sync -> per-wave load/compute pipelining), then pass1 S = Q @ KCP^T
// and pass2 O~ = P @ KC via v_wmma_f32_16x16x4_f32.
// kv LDS rows: 512 dwords with +34 pad after each 256 -> stride 580 (==4 mod
// 64 banks); column c lives at c + 34*(c>=256). pe rows: 64 + 4 -> stride 68.
// ---------------------------------------------------------------------------
__global__ void attn_chunk_kernel(const float* __restrict__ kvc,
                                  const float* __restrict__ pec,
                                  const float* __restrict__ qcat,
                                  const float* __restrict__ nkv,
                                  const float* __restrict__ npe,
                                  float* __restrict__ mbuf,
                                  float* __restrict__ lbuf,
                                  float* __restrict__ obuf) {
  extern __shared__ float sm[];
  float* s_kv = sm + L_KV;
  float* s_pe = sm + L_PE;
  float* s_q  = sm + L_Q;
  float* s_p  = sm + L_P;
  const int tid = threadIdx.x;
  const int wave = tid >> 5, lane = tid & 31;
  const int ch = blockIdx.x, b = blockIdx.y;
  const int t0 = ch * CHUNK;
  const unsigned lds0 = __builtin_amdgcn_groupstaticsize();

  // ---- each pass-1 wave DMAs its own 16-row tile (async, per-wave) ----
  const int rowbase = t0 + wave * 16;            // meaningful for wave < 6
  const int avail = (SEQ - 1) - rowbase;         // rows present in the cache
  if (wave < 6) {
    if (avail > 0) {
      tdm_load_2d(lds0 + (unsigned)(L_KV + wave * 16 * KSTR) * 4u,
                  kvc + ((size_t)b * SEQ + rowbase) * 512,
                  512u, (unsigned)avail, 16u, /*pad: 34 dw per 256*/ 7u, 33u);
      tdm_load_2d(lds0 + (unsigned)(L_PE + wave * 16 * ESTR) * 4u,
                  pec + ((size_t)b * SEQ + rowbase) * 64,
                  64u, (unsigned)avail, 16u, /*pad: 4 dw per 64*/ 5u, 3u);
    } else {
      // tile entirely past the cache: zero-fill (finite for pass2's 0*x)
      for (int i = lane; i < 16 * KSTR; i += 32) s_kv[wave * 16 * KSTR + i] = 0.f;
      for (int i = lane; i < 16 * ESTR; i += 32) s_pe[wave * 16 * ESTR + i] = 0.f;
    }
  }

  // ---- overlap: cooperative load of q_cat[b] (16 x 576) into LDS ----
  for (int q4 = tid; q4 < 16 * 144; q4 += 256) {
    const int row = q4 / 144, j = q4 % 144;
    const float4 v =
        *(const float4*)(qcat + ((size_t)(b * 16 + row)) * KDIM + j * 4);
    *(float4*)(s_q + row * KSTR + j * 4) = v;
  }
  __syncthreads();   // q ready; TDM tiles may still be in flight

  const int lo16 = lane & 15;
  const int koff = (lane < 16) ? 0 : 2;
  const int rbase = (lane < 16) ? 0 : 8;

  // ---- pass 1: each wave waits only for ITS tile, then scores it ----
  if (wave < 6) {
    __builtin_amdgcn_s_wait_tensorcnt(0);

    // patch in the new position-8192 row if it falls in this tile
    if (rowbase <= SEQ - 1 && SEQ - 1 < rowbase + 16) {
      const int r = (SEQ - 1) - t0;    // row within the chunk
      for (int i = lane; i < KDIM; i += 32) {
        if (i < 512) s_kv[r * KSTR + i + ((i >> 8) * 34)] = nkv[(size_t)b * 512 + i];
        else         s_pe[r * ESTR + (i - 512)] = npe[(size_t)b * 64 + (i - 512)];
      }
    }

    v8f acc = {};
    const float* qrow = s_q + lo16 * KSTR;                    // A: m = lane
    const float* krow = s_kv + (wave * 16 + lo16) * KSTR;     // B: n = lane
    const float* erow = s_pe + (wave * 16 + lo16) * ESTR;
    for (int k0 = 0; k0 < 256; k0 += 4) {                     // kv cols 0..255
      v2f a, bb;
      a.x = qrow[k0 + koff];  a.y = qrow[k0 + koff + 1];
      bb.x = krow[k0 + koff]; bb.y = krow[k0 + koff + 1];
      acc = wmma4(a, bb, acc);
    }
    for (int k0 = 256; k0 < 512; k0 += 4) {                   // kv cols 256..511
      v2f a, bb;
      a.x = qrow[k0 + koff];       a.y = qrow[k0 + koff + 1];
      bb.x = krow[k0 + 34 + koff]; bb.y = krow[k0 + 34 + koff + 1];
      acc = wmma4(a, bb, acc);
    }
    for (int k0 = 0; k0 < 64; k0 += 4) {                      // pe cols
      v2f a, bb;
      a.x = qrow[512 + k0 + koff]; a.y = qrow[512 + k0 + koff + 1];
      bb.x = erow[k0 + koff];      bb.y = erow[k0 + koff + 1];
      acc = wmma4(a, bb, acc);
    }
    const int col = wave * 16 + lo16;
    for (int j = 0; j < 8; ++j) s_p[(rbase + j) * PSTR + col] = acc[j];
  }
  __syncthreads();

  // ---- chunk-local softmax partials (16 heads, serial over <=96 t) ----
  const int vlen = (SEQ - t0 < CHUNK) ? (SEQ - t0) : CHUNK;
  if (tid < 16) {
    const int h = tid;
    float m = -1e30f;
    for (int t = 0; t < vlen; ++t) m = fmaxf(m, s_p[h * PSTR + t]);
    float l = 0.f;
    for (int t = 0; t < vlen; ++t) {
      const float pr = __expf(SCALE * (s_p[h * PSTR + t] - m));
      s_p[h * PSTR + t] = pr;
      l += pr;
    }
    for (int t = vlen; t < CHUNK; ++t) s_p[h * PSTR + t] = 0.f;
    mbuf[((size_t)b * NCH + ch) * 16 + h] = m;
    lbuf[((size_t)b * NCH + ch) * 16 + h] = l;
  }
  __syncthreads();

  // ---- pass 2: O~[16h x 512c] = P[16x96] @ KC[96x512] ----
  {
    const float* prow = s_p + lo16 * PSTR;                    // A: m = lane
    const size_t obase = ((size_t)(b * NCH + ch) * 16) * 512;
    for (int u = 0; u < 4; ++u) {
      const int cbase = wave * 64 + u * 16;
      const int foff = (cbase >= 256) ? 34 : 0;               // TDM pad skew
      v8f acc = {};
      for (int k0 = 0; k0 < CHUNK; k0 += 4) {
        v2f a, bb;
        a.x = prow[k0 + koff];  a.y = prow[k0 + koff + 1];
        bb.x = s_kv[(k0 + koff) * KSTR + cbase + foff + lo16];
        bb.y = s_kv[(k0 + koff + 1) * KSTR + cbase + foff + lo16];
        acc = wmma4(a, bb, acc);
      }
      for (int j = 0; j < 8; ++j)
        obuf[obase + (size_t)(rbase + j) * 512 + cbase + lo16] = acc[j];
    }
  }
}

// ---------------------------------------------------------------------------
// Log-sum-exp combine of flash partials across chunks.  grid (b, h), 128 thr.
// ---------------------------------------------------------------------------
__global__ void combine_kernel(const float* __restrict__ mbuf,
                               const float* __restrict__ lbuf,
                               const float* __restrict__ obuf,
                               float* __restrict__ ohead) {
  const int b = blockIdx.x, h = blockIdx.y, tid = threadIdx.x;
  float M = -1e30f;
  for (int ch = 0; ch < NCH; ++ch)
    M = fmaxf(M, mbuf[((size_t)b * NCH + ch) * 16 + h]);
  float L = 0.f;
  float4 acc = {0.f, 0.f, 0.f, 0.f};
  for (int ch = 0; ch < NCH; ++ch) {
    const float w = __expf(SCALE * (mbuf[((size_t)b * NCH + ch) * 16 + h] - M));
    L += lbuf[((size_t)b * NCH + ch) * 16 + h] * w;
    const float4 v = *(const float4*)(obuf +
        ((size_t)(b * NCH + ch) * 16 + h) * 512 + tid * 4);
    acc.x += v.x * w; acc.y += v.y * w; acc.z += v.z * w; acc.w += v.w * w;
  }
  const float inv = 1.f / L;
  float4 r = {acc.x * inv, acc.y * inv, acc.z * inv, acc.w * inv};
  *(float4*)(ohead + ((size_t)b * 16 + h) * 512 + tid * 4) = r;
}

// ---------------------------------------------------------------------------
// o_proj: o_flat[b, h*128+d] = sum_c ohead[b,h,c] * wkv_b[h*256+128+d, c]
// ---------------------------------------------------------------------------
__global__ void oproj_kernel(const float* __restrict__ ohead,
                             const float* __restrict__ wkv_b,
                             float* __restrict__ oflat) {
  __shared__ float os[512];
  const int b = blockIdx.x, h = blockIdx.y, tid = threadIdx.x;
  for (int i = tid; i < 512; i += 256)
    os[i] = ohead[((size_t)b * 16 + h) * 512 + i];
  __syncthreads();
  const int wave = tid >> 5, lane = tid & 31;
  for (int d = wave * 16; d < wave * 16 + 16; ++d) {
    const float* wr = wkv_b + ((size_t)(h * 256 + 128 + d)) * 512;
    float s = 0.f;
    for (int c = lane; c < 512; c += 32) s += wr[c] * os[c];
    for (int off = 16; off > 0; off >>= 1) s += __shfl_xor(s, off, 32);
    if (lane == 0) oflat[(size_t)b * 2048 + h * 128 + d] = s;
  }
}

// ---------------------------------------------------------------------------
extern "C" void kernel_launch(void* const* d_in, const int* in_sizes, int n_in,
                              void* d_out, int out_size, void* d_ws, size_t ws_size,
                              hipStream_t stream) {
  const float* x     = (const float*)d_in[0];   // (16,1,2048)
  const float* wq    = (const float*)d_in[1];   // (3072,2048)
  const float* wkv_a = (const float*)d_in[2];   // (576,2048)
  const float* kvnw  = (const float*)d_in[3];   // (512,)
  const float* wkv_b = (const float*)d_in[4];   // (4096,512)
  const float* wo    = (const float*)d_in[5];   // (2048,2048)
  const float* kvc   = (const float*)d_in[6];   // (16,8193,512)
  const float* pec   = (const float*)d_in[7];   // (16,8193,64)
  const float* cosv  = (const float*)d_in[8];   // (1,32)
  const float* sinv  = (const float*)d_in[9];   // (1,32)
  (void)in_sizes; (void)n_in; (void)out_size; (void)ws_size;

  float* ws    = (float*)d_ws;
  float* qraw  = ws + OFF_QRAW;
  float* qcat  = ws + OFF_QCAT;
  float* nkv   = ws + OFF_NKV;
  float* npe   = ws + OFF_NPE;
  float* mbuf  = ws + OFF_M;
  float* lbuf  = ws + OFF_L;
  float* obuf  = ws + OFF_O;
  float* ohead = ws + OFF_OHEAD;
  float* oflat = ws + OFF_OFLAT;

  // q = x @ wq^T  (3072 rows), kv_full = x @ wkv_a^T (576 rows)
  gemv2048_kernel<<<dim3(16, 48), 256, 0, stream>>>(x, wq, qraw, 3072, 2048, 3648);
  gemv2048_kernel<<<dim3(16, 9), 256, 0, stream>>>(x, wkv_a, qraw + 3072, 576, 2048, 3648);

  // rope / rms / q_abs
  prep_kernel<<<16, 256, 0, stream>>>(qraw, kvnw, wkv_b, cosv, sinv, qcat, nkv, npe);

  // flash-decoding chunks (WMMA + per-wave TDM pipeline), 292,352 B LDS/WG
  attn_chunk_kernel<<<dim3(NCH, 16), 256, L_TOT * 4, stream>>>(
      kvc, pec, qcat, nkv, npe, mbuf, lbuf, obuf);

  // combine partials, project through wkv_b value part, then wo
  combine_kernel<<<dim3(16, 16), 128, 0, stream>>>(mbuf, lbuf, obuf, ohead);
  oproj_kernel<<<dim3(16, 16), 256, 0, stream>>>(ohead, wkv_b, oflat);
  gemv2048_kernel<<<dim3(16, 32), 256, 0, stream>>>(oflat, wo, (float*)d_out,
                                                    2048, 2048, 2048);
}